// GAT_33672543600972
// MI455X (gfx1250) — compile-verified
//
#include <hip/hip_runtime.h>
#include <math.h>

typedef float v2f __attribute__((ext_vector_type(2)));
typedef float v8f __attribute__((ext_vector_type(8)));

#define LRELU 0.2f

// ---- order-preserving float<->uint encoding for atomic max on floats ----
__device__ __forceinline__ unsigned encOrd(float f){
  unsigned u = __float_as_uint(f);
  return (u & 0x80000000u) ? ~u : (u | 0x80000000u);
}
__device__ __forceinline__ float decOrd(unsigned u){
  return (u & 0x80000000u) ? __uint_as_float(u ^ 0x80000000u) : __uint_as_float(~u);
}

__device__ __forceinline__ void atomAddF(float* p, float v){
  unsafeAtomicAdd(p, v);   // hardware global_atomic_add_f32
}

// ------------------------- fills -------------------------
__global__ void k_fill_f32(float* __restrict__ p, float v, int n){
  int i = blockIdx.x*blockDim.x + threadIdx.x;
  if (i < n) p[i] = v;
}
__global__ void k_fill_u32(unsigned* __restrict__ p, unsigned v, int n){
  int i = blockIdx.x*blockDim.x + threadIdx.x;
  if (i < n) p[i] = v;
}

// ---------- pack W[K x Fout] into WMMA-fragment order, zero-padded to FoutPad ----------
// Packed element i = (t, col) with t = kk*2 + half:  float2{ W[kk*4+half*2][col], W[kk*4+half*2+1][col] }
__global__ void k_pack_w(const float* __restrict__ W, float2* __restrict__ Wp,
                         int K, int Fout, int FoutPad){
  int i = blockIdx.x*blockDim.x + threadIdx.x;
  int total = (K/4) * 2 * FoutPad;
  if (i >= total) return;
  int col = i % FoutPad;
  int t   = i / FoutPad;
  int ka  = (t >> 1)*4 + (t & 1)*2;
  float2 v; v.x = 0.f; v.y = 0.f;
  if (col < Fout){
    v.x = W[(size_t)ka    *Fout + col];
    v.y = W[(size_t)(ka+1)*Fout + col];
  }
  Wp[i] = v;
}

// ---------- H[M x Fout] = X[M x KK] * W[KK x Fout], fp32 WMMA 16x16x4 ----------
// One wave computes a 16-row x (TN*16)-col strip. Requires M % 16 == 0, KK % 4 == 0.
// Packed weights are staged in LDS once per block; B fragments come from ds_load_b64,
// A is a single coalesced global b64 per k-step. K loop fully unrolled (immediate offsets).
template<int TN, int KK>
__global__ void __launch_bounds__(256)
k_gemm_wmma_f32(const float* __restrict__ X, const float2* __restrict__ Wp,
                float* __restrict__ Hm, int M, int Fout){
  const int FoutPad = TN * 16;
  const int ROWS = (KK/4) * 2;                       // K-fragment rows (64 for K=128)
  __shared__ float2 sW[ROWS * FoutPad];              // 24KB (TN=3) / 64KB (TN=8)
  for (int i = threadIdx.x; i < ROWS*FoutPad; i += blockDim.x)
    sW[i] = Wp[i];
  __syncthreads();                                   // all waves participate, then guard

  int wave = (int)((blockIdx.x * blockDim.x + threadIdx.x) >> 5);
  int lane = threadIdx.x & 31;
  int tilesM = M >> 4;
  if (wave >= tilesM) return;                        // wave-uniform: EXEC all-ones inside
  int half = lane >> 4;                              // 0: lanes 0-15, 1: lanes 16-31
  int l16  = lane & 15;
  // A: lane holds row m = l16, K pair (k0 + half*2, +1) -> float2 index kk*2 + half
  const float2* xr = (const float2*)(X + (size_t)(wave*16 + l16) * KK);
  const float2* sw = sW + l16;

  v8f acc[TN];
#pragma unroll
  for (int t = 0; t < TN; ++t) acc[t] = (v8f){};

#pragma unroll
  for (int kk = 0; kk < (KK >> 2); ++kk){
    float2 af = xr[kk*2 + half];
    v2f a; a.x = af.x; a.y = af.y;
    const float2* wrow = sw + (kk*2 + half) * FoutPad;
    float2 bf[TN];
#pragma unroll
    for (int t = 0; t < TN; ++t) bf[t] = wrow[t*16];  // batched ds_load_b64, one wait
#pragma unroll
    for (int t = 0; t < TN; ++t){
      v2f b; b.x = bf[t].x; b.y = bf[t].y;
      acc[t] = __builtin_amdgcn_wmma_f32_16x16x4_f32(false, a, false, b, (short)0, acc[t], false, false);
    }
  }
  // store: lanes 0-15 hold rows v, lanes 16-31 rows v+8 (C/D layout)
  int rbase = wave*16 + half*8;
#pragma unroll
  for (int t = 0; t < TN; ++t){
    int col = t*16 + l16;
    if (col < Fout){
#pragma unroll
      for (int v = 0; v < 8; ++v)
        Hm[(size_t)(rbase + v)*Fout + col] = acc[t][v];
    }
  }
}

// ---------- asrc[n,h] = <h[n,h,:], a_src[h,:]>, adst likewise. wave per (n,h) ----------
__global__ void k_attn_scores(const float* __restrict__ Hm, const float* __restrict__ aS,
                              const float* __restrict__ aD, float* __restrict__ asrc,
                              float* __restrict__ adst, int N, int Hh, int C){
  int wave = (int)((blockIdx.x*blockDim.x + threadIdx.x) >> 5);
  int lane = threadIdx.x & 31;
  if (wave >= N*Hh) return;
  int n = wave / Hh, h = wave % Hh;
  const float* row = Hm + (size_t)n*Hh*C + (size_t)h*C;
  const float* s0 = aS + h*C;
  const float* d0 = aD + h*C;
  float s = 0.f, d = 0.f;
  for (int c = lane; c < C; c += 32){ float v = row[c]; s += v*s0[c]; d += v*d0[c]; }
#pragma unroll
  for (int off = 16; off > 0; off >>= 1){
    s += __shfl_down(s, off, 32);
    d += __shfl_down(d, off, 32);
  }
  if (lane == 0){ asrc[wave] = s; adst[wave] = d; }
}

// ---------- segment max over incoming edges (thread per (edge,head)) ----------
template<int HH>
__global__ void k_edge_max(const int* __restrict__ src, const int* __restrict__ dst,
                           int E, int N, const float* __restrict__ asrc,
                           const float* __restrict__ adst, unsigned* __restrict__ emax){
  int idx = blockIdx.x*blockDim.x + threadIdx.x;
  int total = (E + N) * HH;
  if (idx >= total) return;
  int e = idx / HH, h = idx - e*HH;
  int s, d;
  if (e < E){ s = src[e]; d = dst[e]; } else { s = d = e - E; }   // self loops
  float x = asrc[s*HH + h] + adst[d*HH + h];
  x = x > 0.f ? x : LRELU * x;
  atomicMax(&emax[d*HH + h], encOrd(x));
}

// ---------- segment sum of exp(e - emax) ----------
template<int HH>
__global__ void k_edge_sum(const int* __restrict__ src, const int* __restrict__ dst,
                           int E, int N, const float* __restrict__ asrc,
                           const float* __restrict__ adst, const unsigned* __restrict__ emax,
                           float* __restrict__ denom){
  int idx = blockIdx.x*blockDim.x + threadIdx.x;
  int total = (E + N) * HH;
  if (idx >= total) return;
  int e = idx / HH, h = idx - e*HH;
  int s, d;
  if (e < E){ s = src[e]; d = dst[e]; } else { s = d = e - E; }
  float x = asrc[s*HH + h] + adst[d*HH + h];
  x = x > 0.f ? x : LRELU * x;
  float ex = __expf(x - decOrd(emax[d*HH + h]));
  atomAddF(&denom[d*HH + h], ex);
}

// ---------- scatter: acc[dst] += alpha * h[src]. wave per edge, float4 per lane ----------
template<int HH, int CC>
__global__ void k_edge_scatter(const int* __restrict__ src, const int* __restrict__ dst,
                               int E, int N, const float* __restrict__ asrc,
                               const float* __restrict__ adst, const unsigned* __restrict__ emax,
                               const float* __restrict__ denom, const float* __restrict__ Hm,
                               float* __restrict__ acc){
  int wave = (int)((blockIdx.x*blockDim.x + threadIdx.x) >> 5);
  int lane = threadIdx.x & 31;
  int EP = E + N;
  if (wave >= EP) return;
  int s, d;
  if (wave < E){ s = src[wave]; d = dst[wave]; } else { s = d = wave - E; }
  float alpha[HH];
#pragma unroll
  for (int h = 0; h < HH; ++h){
    float x = asrc[s*HH + h] + adst[d*HH + h];
    x = x > 0.f ? x : LRELU * x;
    alpha[h] = __expf(x - decOrd(emax[d*HH + h])) / denom[d*HH + h];
  }
  const int HC = HH * CC;                          // 128 or 40, both multiples of 4
  const float4* hs4 = reinterpret_cast<const float4*>(Hm + (size_t)s * HC);
  float* ao = acc + (size_t)d * HC;
#pragma unroll
  for (int q0 = 0; q0 < HC/4; q0 += 32){
    int q = q0 + lane;
    if (q*4 < HC){
      float4 v = hs4[q];                           // global_load_b128 gather (L2-resident)
      float al = alpha[(q*4) / CC];                // compile-time division
      atomAddF(ao + q*4 + 0, v.x * al);
      atomAddF(ao + q*4 + 1, v.y * al);
      atomAddF(ao + q*4 + 2, v.z * al);
      atomAddF(ao + q*4 + 3, v.w * al);
    }
  }
}

// ---------- out = elu(acc + bias), in place ----------
__global__ void k_bias_elu(float* __restrict__ acc, const float* __restrict__ b, int N, int F){
  int i = blockIdx.x*blockDim.x + threadIdx.x;
  if (i >= N*F) return;
  float v = acc[i] + b[i % F];
  acc[i] = v > 0.f ? v : expm1f(v);
}

// ---------- final: softmax(acc + b2) over 40 classes, thread per node ----------
__global__ void k_bias_softmax40(const float* __restrict__ acc, const float* __restrict__ b,
                                 float* __restrict__ out, int N){
  int n = blockIdx.x*blockDim.x + threadIdx.x;
  if (n >= N) return;
  float v[40]; float m = -3.4e38f;
#pragma unroll
  for (int c = 0; c < 40; ++c){ v[c] = acc[(size_t)n*40 + c] + b[c]; m = fmaxf(m, v[c]); }
  float ssum = 0.f;
#pragma unroll
  for (int c = 0; c < 40; ++c){ v[c] = __expf(v[c] - m); ssum += v[c]; }
  float inv = 1.f / ssum;
#pragma unroll
  for (int c = 0; c < 40; ++c) out[(size_t)n*40 + c] = v[c] * inv;
}

// ======================= host-side layer driver =======================
static inline int cdiv(long long a, long long b){ return (int)((a + b - 1) / b); }

static void run_gat_layer(const float* X, int Fin, const float* W, int Hh, int C,
                          const float* aS, const float* aD,
                          float* hbuf, float* acc, float2* wpack,
                          const int* src, const int* dst, int E, int N,
                          float* asrc, float* adst, unsigned* emax, float* denom,
                          hipStream_t stream){
  const int B = 256;
  int Fout = Hh * C;
  int tilesN = (Fout + 15) / 16;
  int FoutPad = tilesN * 16;
  // 0) pack/pad W into fragment order
  int packElems = (Fin/4) * 2 * FoutPad;
  k_pack_w<<<cdiv(packElems, B), B, 0, stream>>>(W, wpack, Fin, Fout, FoutPad);
  // 1) h = X @ W (WMMA); one wave per 16-row strip (Fin is 128 in all layers)
  long long gthreads = (long long)(N/16) * 32;
  if (tilesN == 8)
    k_gemm_wmma_f32<8,128><<<cdiv(gthreads, B), B, 0, stream>>>(X, wpack, hbuf, N, Fout);
  else
    k_gemm_wmma_f32<3,128><<<cdiv(gthreads, B), B, 0, stream>>>(X, wpack, hbuf, N, Fout);
  // 2) attention dots
  k_attn_scores<<<cdiv((long long)N*Hh*32, B), B, 0, stream>>>(hbuf, aS, aD, asrc, adst, N, Hh, C);
  // 3) init segment state + accumulator (input X is dead now; acc may alias it)
  k_fill_u32<<<cdiv((long long)N*Hh, B), B, 0, stream>>>(emax, 0u, N*Hh);   // encOrd-minimum
  k_fill_f32<<<cdiv((long long)N*Hh, B), B, 0, stream>>>(denom, 0.f, N*Hh);
  k_fill_f32<<<cdiv((long long)N*Fout, B), B, 0, stream>>>(acc, 0.f, N*Fout);
  // 4) segment softmax + scatter
  long long eph = (long long)(E + N) * Hh;
  long long sth = (long long)(E + N) * 32;
  if (Hh == 2){
    k_edge_max<2><<<cdiv(eph, B), B, 0, stream>>>(src, dst, E, N, asrc, adst, emax);
    k_edge_sum<2><<<cdiv(eph, B), B, 0, stream>>>(src, dst, E, N, asrc, adst, emax, denom);
    k_edge_scatter<2,64><<<cdiv(sth, B), B, 0, stream>>>(src, dst, E, N, asrc, adst,
                                                         emax, denom, hbuf, acc);
  } else {
    k_edge_max<1><<<cdiv(eph, B), B, 0, stream>>>(src, dst, E, N, asrc, adst, emax);
    k_edge_sum<1><<<cdiv(eph, B), B, 0, stream>>>(src, dst, E, N, asrc, adst, emax, denom);
    k_edge_scatter<1,40><<<cdiv(sth, B), B, 0, stream>>>(src, dst, E, N, asrc, adst,
                                                         emax, denom, hbuf, acc);
  }
}

extern "C" void kernel_launch(void* const* d_in, const int* in_sizes, int n_in,
                              void* d_out, int out_size, void* d_ws, size_t ws_size,
                              hipStream_t stream) {
  (void)n_in; (void)out_size; (void)ws_size;
  const float* x   = (const float*)d_in[0];
  const int*   ei  = (const int*)d_in[1];
  const float* W0  = (const float*)d_in[2];
  const float* as0 = (const float*)d_in[3];
  const float* ad0 = (const float*)d_in[4];
  const float* b0  = (const float*)d_in[5];
  const float* W1  = (const float*)d_in[6];
  const float* as1 = (const float*)d_in[7];
  const float* ad1 = (const float*)d_in[8];
  const float* b1  = (const float*)d_in[9];
  const float* W2  = (const float*)d_in[10];
  const float* as2 = (const float*)d_in[11];
  const float* ad2 = (const float*)d_in[12];
  const float* b2  = (const float*)d_in[13];

  const int NFEAT = 128, NHID = 64, HEADS = 2, NCLASS = 40;
  int N = in_sizes[0] / NFEAT;       // 100000 (divisible by 16)
  int E = in_sizes[1] / 2;           // 1600000
  const int* src = ei;
  const int* dst = ei + E;

  // workspace layout
  char* ws = (char*)d_ws;
  size_t big = (size_t)N * (HEADS*NHID) * sizeof(float);   // 51.2 MB
  float*    bufA  = (float*)(ws);                          // h of current layer
  float*    bufB  = (float*)(ws + big);                    // layer output (ping-pong)
  float*    acc2  = (float*)(ws + 2*big);                  // N*40 final accumulator
  char*     small = ws + 2*big + (size_t)N * NCLASS * sizeof(float);
  float*    asrc  = (float*)(small);
  float*    adst  = asrc  + (size_t)N * HEADS;
  unsigned* emax  = (unsigned*)(adst + (size_t)N * HEADS);
  float*    denom = (float*)(emax + (size_t)N * HEADS);
  float2*   wpack = (float2*)(denom + (size_t)N * HEADS);  // <= 64 KB packed weights

  const int B = 256;

  // Layer 0: x[N,128] -> bufB[N,128], ELU
  run_gat_layer(x, NFEAT, W0, HEADS, NHID, as0, ad0, bufA, bufB, wpack,
                src, dst, E, N, asrc, adst, emax, denom, stream);
  k_bias_elu<<<cdiv((long long)N*HEADS*NHID, B), B, 0, stream>>>(bufB, b0, N, HEADS*NHID);

  // Layer 1: bufB -> bufB (GEMM reads bufB into bufA first, then acc reuses bufB)
  run_gat_layer(bufB, HEADS*NHID, W1, HEADS, NHID, as1, ad1, bufA, bufB, wpack,
                src, dst, E, N, asrc, adst, emax, denom, stream);
  k_bias_elu<<<cdiv((long long)N*HEADS*NHID, B), B, 0, stream>>>(bufB, b1, N, HEADS*NHID);

  // Layer 2: bufB -> acc2[N,40] (1 head, mean over 1 head == identity)
  run_gat_layer(bufB, HEADS*NHID, W2, 1, NCLASS, as2, ad2, bufA, acc2, wpack,
                src, dst, E, N, asrc, adst, emax, denom, stream);

  k_bias_softmax40<<<cdiv(N, B), B, 0, stream>>>(acc2, b2, (float*)d_out, N);
}